// GATODEFunc_44779329028551
// MI455X (gfx1250) — compile-verified
//
#include <hip/hip_runtime.h>
#include <math.h>

#define NNODES 50000        // divisible by 16 -> no GEMM tail guards needed
#define NEDGES 800000
#define HID    64
#define HC     256          // H * C
#define NHEADS 4
#define LAYERS 2
#define NEG_SLOPE 0.2f
#define LN_EPS 1e-5f

typedef float v2f __attribute__((ext_vector_type(2)));
typedef float v8f __attribute__((ext_vector_type(8)));

// ---------------------------------------------------------------------------
// x = z @ W^T   z:[N,64]  W:[256,64]  x:[N,256]
// block = 512 threads = 16 waves; block owns a 16-row strip (staged in LDS),
// wave w computes the 16-col tile starting at 16*w using f32 WMMA 16x16x4.
// N % 16 == 0, so no bounds guards anywhere (straight-line stores).
// ---------------------------------------------------------------------------
__global__ __launch_bounds__(512)
void gemm_node_kernel(const float* __restrict__ z, const float* __restrict__ W,
                      float* __restrict__ x) {
    __shared__ float Atile[16 * 64];          // 4 KB
    const int row0 = blockIdx.x * 16;
    const int tid  = threadIdx.x;

    // cooperative load of the 16x64 A strip (2 floats per thread)
    for (int i = tid; i < 16 * 64; i += 512) {
        Atile[i] = z[(size_t)(row0 + (i >> 6)) * 64 + (i & 63)];
    }
    __syncthreads();

    const int wave  = tid >> 5;               // 0..15 -> column tile
    const int lane  = tid & 31;
    const int m     = lane & 15;              // row within tile (A), col for B
    const int khalf = lane >> 4;              // 0 or 1
    const int ncol  = (wave << 4) + m;        // global output column for B loads

    v8f acc = {};
    #pragma unroll
    for (int k0 = 0; k0 < 64; k0 += 4) {
        v2f a, b;
        // A 16x4 layout: VGPR0 -> K = k0 + 2*khalf, VGPR1 -> K+1 ; M = lane&15
        a.x = Atile[m * 64 + k0 + 2 * khalf + 0];
        a.y = Atile[m * 64 + k0 + 2 * khalf + 1];
        // B 4x16 layout mirrors A: B[k][n] = W[n*64 + k]
        b.x = W[ncol * 64 + k0 + 2 * khalf + 0];
        b.y = W[ncol * 64 + k0 + 2 * khalf + 1];
        acc = __builtin_amdgcn_wmma_f32_16x16x4_f32(
            /*neg_a=*/false, a, /*neg_b=*/false, b,
            /*c_mod=*/(short)0, acc, /*reuse_a=*/false, /*reuse_b=*/false);
    }

    // D layout: VGPR i -> M = i + 8*(lane>=16), N = lane&15
    const int colg  = (wave << 4) + (lane & 15);
    const int rbase = row0 + ((lane < 16) ? 0 : 8);
    #pragma unroll
    for (int i = 0; i < 8; ++i) {
        x[(size_t)(rbase + i) * HC + colg] = acc[i];
    }
}

// ---------------------------------------------------------------------------
// a_src[n,h] = sum_c x[n,h,c]*att_src[h,c] ; same for a_dst. 1 block per node.
// ---------------------------------------------------------------------------
__global__ __launch_bounds__(256)
void attn_node_kernel(const float* __restrict__ x,
                      const float* __restrict__ att_src,
                      const float* __restrict__ att_dst,
                      float* __restrict__ a_src, float* __restrict__ a_dst) {
    __shared__ float s1[256], s2[256];
    const int n   = blockIdx.x;
    const int tid = threadIdx.x;                  // tid = h*64 + c
    float xv = x[(size_t)n * HC + tid];
    s1[tid] = xv * att_src[tid];
    s2[tid] = xv * att_dst[tid];
    __syncthreads();
    for (int off = 32; off > 0; off >>= 1) {
        if ((tid & 63) < off) { s1[tid] += s1[tid + off]; s2[tid] += s2[tid + off]; }
        __syncthreads();
    }
    if ((tid & 63) == 0) {
        a_src[n * NHEADS + (tid >> 6)] = s1[tid];
        a_dst[n * NHEADS + (tid >> 6)] = s2[tid];
    }
}

// ---------------------------------------------------------------------------
// attM[h][d] = sum_c lin_edge_W[h*64+c, d] * att_edge[h*64+c]   (12 values)
// ---------------------------------------------------------------------------
__global__ void edge_mat_kernel(const float* __restrict__ lin_edge_W,
                                const float* __restrict__ att_edge,
                                float* __restrict__ attM) {
    int t = threadIdx.x;
    if (t < 12) {
        int d = t % 3, h = t / 3;
        float s = 0.f;
        for (int c = 0; c < 64; ++c)
            s += lin_edge_W[(h * 64 + c) * 3 + d] * att_edge[h * 64 + c];
        attM[h * 3 + d] = s;
    }
}

// vectorized init: msg zeroed as float4; amax/denom per scalar
__global__ __launch_bounds__(256)
void init_kernel(float* __restrict__ amax, float* __restrict__ denom,
                 float4* __restrict__ msg4) {
    int i = blockIdx.x * blockDim.x + threadIdx.x;
    if (i < NNODES * NHEADS) { amax[i] = -INFINITY; denom[i] = 0.f; }
    if (i < NNODES * HC / 4) msg4[i] = make_float4(0.f, 0.f, 0.f, 0.f);
}

__device__ inline void atomicMaxFloat(float* addr, float value) {
    if (value >= 0.f) atomicMax((int*)addr, __float_as_int(value));
    else              atomicMin((unsigned int*)addr, __float_as_uint(value));
}

// ---------------------------------------------------------------------------
// alpha = leaky_relu(a_src[src] + a_dst[dst] + edge_attr @ attM) ; segment max
// ---------------------------------------------------------------------------
__global__ __launch_bounds__(256)
void edge_alpha_kernel(const int* __restrict__ ei, const float* __restrict__ ea,
                       const float* __restrict__ a_src, const float* __restrict__ a_dst,
                       const float* __restrict__ attM,
                       float* __restrict__ alpha, float* __restrict__ amax) {
    int i = blockIdx.x * blockDim.x + threadIdx.x;
    if (i >= NEDGES * NHEADS) return;
    int e = i >> 2, h = i & 3;
    int s = ei[e];
    int d = ei[NEDGES + e];
    float ae = ea[e * 3 + 0] * attM[h * 3 + 0]
             + ea[e * 3 + 1] * attM[h * 3 + 1]
             + ea[e * 3 + 2] * attM[h * 3 + 2];
    float a = a_src[s * NHEADS + h] + a_dst[d * NHEADS + h] + ae;
    a = (a > 0.f) ? a : NEG_SLOPE * a;
    alpha[i] = a;
    atomicMaxFloat(&amax[d * NHEADS + h], a);
}

// ---------------------------------------------------------------------------
// ex = exp(alpha - amax[dst]) (in place) ; denom[dst] += ex
// ---------------------------------------------------------------------------
__global__ __launch_bounds__(256)
void edge_exp_kernel(const int* __restrict__ ei, const float* __restrict__ amax,
                     float* __restrict__ alpha_ex, float* __restrict__ denom) {
    int i = blockIdx.x * blockDim.x + threadIdx.x;
    if (i >= NEDGES * NHEADS) return;
    int e = i >> 2, h = i & 3;
    int d = ei[NEDGES + e];
    float am = amax[d * NHEADS + h];
    if (!__builtin_isfinite(am)) am = 0.f;
    float ex = expf(alpha_ex[i] - am);
    alpha_ex[i] = ex;
    atomicAdd(&denom[d * NHEADS + h], ex);
}

// ---------------------------------------------------------------------------
// msg[dst, h, c] += (ex/denom[dst]) * x[src, h, c]
// block = 256 threads = 4 edges x 64 threads; each thread: float4 (b128) load
// of x + 4 f32 atomic adds into L2-resident msg.
// ---------------------------------------------------------------------------
__global__ __launch_bounds__(256)
void scatter_kernel(const int* __restrict__ ei, const float* __restrict__ x,
                    const float* __restrict__ ex, const float* __restrict__ denom,
                    float* __restrict__ msg) {
    const int tid = threadIdx.x;
    const int e   = blockIdx.x * 4 + (tid >> 6);      // edge handled by this 64-group
    const int t   = tid & 63;                          // 0..63 -> channels 4t..4t+3
    const int h   = t >> 4;                            // (4t)/64
    const int s   = ei[e];
    const int d   = ei[NEDGES + e];
    const float w = ex[e * NHEADS + h] / (denom[d * NHEADS + h] + 1e-16f);
    const float4 xv = *(const float4*)(x + (size_t)s * HC + 4 * t);
    float* mp = msg + (size_t)d * HC + 4 * t;
    atomicAdd(mp + 0, w * xv.x);
    atomicAdd(mp + 1, w * xv.y);
    atomicAdd(mp + 2, w * xv.z);
    atomicAdd(mp + 3, w * xv.w);
}

// ---------------------------------------------------------------------------
// out = mean_h(msg) + bias ; LayerNorm ; SiLU.  1 block (64 thr) per node.
// ---------------------------------------------------------------------------
__global__ __launch_bounds__(64)
void finalize_kernel(const float* __restrict__ msg, const float* __restrict__ bias,
                     const float* __restrict__ gamma, const float* __restrict__ beta,
                     float* __restrict__ zout) {
    __shared__ float red[64];
    const int n = blockIdx.x, c = threadIdx.x;
    const float* m = msg + (size_t)n * HC;
    float out = 0.25f * (m[c] + m[64 + c] + m[128 + c] + m[192 + c]) + bias[c];

    red[c] = out; __syncthreads();
    for (int off = 32; off > 0; off >>= 1) {
        if (c < off) red[c] += red[c + off];
        __syncthreads();
    }
    float mu = red[0] * (1.f / 64.f);
    __syncthreads();
    float dc = out - mu;
    red[c] = dc * dc; __syncthreads();
    for (int off = 32; off > 0; off >>= 1) {
        if (c < off) red[c] += red[c + off];
        __syncthreads();
    }
    float var = red[0] * (1.f / 64.f);
    float ln = dc * rsqrtf(var + LN_EPS) * gamma[c] + beta[c];
    zout[(size_t)n * HID + c] = ln / (1.f + expf(-ln));   // SiLU
}

// ---------------------------------------------------------------------------
extern "C" void kernel_launch(void* const* d_in, const int* in_sizes, int n_in,
                              void* d_out, int out_size, void* d_ws, size_t ws_size,
                              hipStream_t stream) {
    const float* h0          = (const float*)d_in[1];
    const int*   ei          = (const int*)d_in[2];    // [2,E]
    const float* ea          = (const float*)d_in[3];  // [E,3]
    const float* lin_W       = (const float*)d_in[4];  // [L,256,64]
    const float* lin_edge_W  = (const float*)d_in[5];  // [L,256,3]
    const float* att_src     = (const float*)d_in[6];  // [L,256]
    const float* att_dst     = (const float*)d_in[7];
    const float* att_edge    = (const float*)d_in[8];
    const float* bias        = (const float*)d_in[9];  // [L,64]
    const float* ln_gamma    = (const float*)d_in[10];
    const float* ln_beta     = (const float*)d_in[11];
    float* out = (float*)d_out;

    // workspace layout (x and msg are L2-resident at 51.2 MB each; ~131 MB total)
    char* p = (char*)d_ws;
    auto take = [&](size_t bytes) { char* q = p; p += (bytes + 255) & ~(size_t)255; return (void*)q; };
    float* x     = (float*)take((size_t)NNODES * HC * 4);
    float* msg   = (float*)take((size_t)NNODES * HC * 4);
    float* alpha = (float*)take((size_t)NEDGES * NHEADS * 4);
    float* asrc  = (float*)take((size_t)NNODES * NHEADS * 4);
    float* adst  = (float*)take((size_t)NNODES * NHEADS * 4);
    float* amax  = (float*)take((size_t)NNODES * NHEADS * 4);
    float* denom = (float*)take((size_t)NNODES * NHEADS * 4);
    float* zbuf  = (float*)take((size_t)NNODES * HID * 4);
    float* attM  = (float*)take(16 * 4);

    for (int l = 0; l < LAYERS; ++l) {
        const float* zin = (l == 0) ? h0 : zbuf;
        float* zout = (l == LAYERS - 1) ? out : zbuf;

        gemm_node_kernel<<<NNODES / 16, 512, 0, stream>>>(
            zin, lin_W + (size_t)l * HC * HID, x);
        attn_node_kernel<<<NNODES, 256, 0, stream>>>(
            x, att_src + l * HC, att_dst + l * HC, asrc, adst);
        edge_mat_kernel<<<1, 16, 0, stream>>>(
            lin_edge_W + (size_t)l * HC * 3, att_edge + l * HC, attM);
        init_kernel<<<(NNODES * HC / 4 + 255) / 256, 256, 0, stream>>>(
            amax, denom, (float4*)msg);
        edge_alpha_kernel<<<(NEDGES * NHEADS + 255) / 256, 256, 0, stream>>>(
            ei, ea, asrc, adst, attM, alpha, amax);
        edge_exp_kernel<<<(NEDGES * NHEADS + 255) / 256, 256, 0, stream>>>(
            ei, amax, alpha, denom);
        scatter_kernel<<<NEDGES / 4, 256, 0, stream>>>(ei, x, alpha, denom, msg);
        finalize_kernel<<<NNODES, 64, 0, stream>>>(
            msg, bias + l * HID, ln_gamma + l * HID, ln_beta + l * HID, zout);
    }
}